// CrossScan_34059090657945
// MI455X (gfx1250) — compile-verified
//
#include <hip/hip_runtime.h>
#include <hip/hip_bf16.h>
#include <stdint.h>

// ---------------- problem constants ----------------
constexpr int BB   = 16;    // batch
constexpr int DD_  = 192;   // feature dim
constexpr int NN   = 1024;  // H*W
constexpr int KNN  = 16;    // knn
constexpr int TK   = 4;     // kept eigenvectors
constexpr int NV   = 4;     // iterated subspace size (null vector handled analytically)
constexpr int NITER = 48;   // power-iteration steps

typedef float v2f __attribute__((ext_vector_type(2)));
typedef float v8f __attribute__((ext_vector_type(8)));

// ---------------- workspace layout (floats) ----------------
constexpr size_t offX   = 0;                               // BB*NN*DD_
constexpr size_t offSQ  = offX  + (size_t)BB*NN*DD_;       // BB*NN
constexpr size_t offSig = offSQ + (size_t)BB*NN;           // 16 (accum + inv2s2)
constexpr size_t offD2  = offSig + 16;                     // BB*NN*NN (clipped d^2)
constexpr size_t offA   = offD2 + (size_t)BB*NN*NN;        // BB*NN*NN (A, then L in place)
constexpr size_t offDis = offA  + (size_t)BB*NN*NN;        // BB*NN  (deg^-1/2)
constexpr size_t offV0  = offDis + (size_t)BB*NN;          // BB*NN  (known null eigvec)
constexpr size_t offU   = offV0 + (size_t)BB*NN;           // BB*NV*NN
constexpr size_t offWv  = offU  + (size_t)BB*NV*NN;        // BB*NV*NN
constexpr size_t offSM  = offWv + (size_t)BB*NV*NN;        // BB*NN*TK (signed embedding)

// ---------------- helpers ----------------
__device__ __forceinline__ float blockDot256(const float* a, const float* b, float* red) {
  float s = 0.f;
  for (int i = threadIdx.x; i < NN; i += 256) s += a[i] * b[i];
  red[threadIdx.x] = s;
  __syncthreads();
  for (int off = 128; off > 0; off >>= 1) {
    if ((int)threadIdx.x < off) red[threadIdx.x] += red[threadIdx.x + off];
    __syncthreads();
  }
  float r = red[0];
  __syncthreads();
  return r;
}

// ---------------- kernels ----------------

// zero adjacency + sigma accumulator (must re-zero every call: determinism)
__global__ void k_zero(float* __restrict__ A, size_t n, float* __restrict__ sig) {
  size_t i = (size_t)blockIdx.x * blockDim.x + threadIdx.x;
  size_t stride = (size_t)gridDim.x * blockDim.x;
  for (; i < n; i += stride) A[i] = 0.f;
  if (blockIdx.x == 0 && threadIdx.x == 0) sig[0] = 0.f;
}

// (B,D,N) -> (B,N,D) row-major
__global__ void k_transpose(const float* __restrict__ F, float* __restrict__ X) {
  size_t idx = (size_t)blockIdx.x * 256 + threadIdx.x;
  size_t total = (size_t)BB * DD_ * NN;
  if (idx >= total) return;
  int n = (int)(idx % NN);
  size_t t = idx / NN;
  int d = (int)(t % DD_);
  int b = (int)(t / DD_);
  X[((size_t)b * NN + n) * DD_ + d] = F[idx];
}

__global__ void k_rownorm(const float* __restrict__ X, float* __restrict__ SQ) {
  int idx = blockIdx.x * 256 + threadIdx.x;
  if (idx >= BB * NN) return;
  const float* xr = X + (size_t)idx * DD_;
  float s = 0.f;
  for (int d = 0; d < DD_; ++d) s += xr[d] * xr[d];
  SQ[idx] = s;
}

// one wave per 16x16 tile of G = X X^T ; d2 = sq_i + sq_j - 2 G ; accumulate sum(dist)
__global__ void k_gram(const float* __restrict__ X, const float* __restrict__ SQ,
                       float* __restrict__ D2, float* __restrict__ sig) {
  int b  = blockIdx.z;
  int r0 = blockIdx.y * 16;
  int c0 = blockIdx.x * 16;
  int lane = threadIdx.x;
  int m  = lane & 15;     // row-in-tile for A, col-in-tile for B/C/D
  int kh = lane >> 4;     // K-half selector (A layout: v0 holds K=0/2, v1 holds K=1/3)
  const float* Xb = X + (size_t)b * NN * DD_;

  v8f acc = {0.f, 0.f, 0.f, 0.f, 0.f, 0.f, 0.f, 0.f};
  for (int k0 = 0; k0 < DD_; k0 += 4) {
    v2f a, bb;
    a.x  = Xb[(size_t)(r0 + m) * DD_ + k0 + 2 * kh];
    a.y  = Xb[(size_t)(r0 + m) * DD_ + k0 + 2 * kh + 1];
    bb.x = Xb[(size_t)(c0 + m) * DD_ + k0 + 2 * kh];
    bb.y = Xb[(size_t)(c0 + m) * DD_ + k0 + 2 * kh + 1];
    acc = __builtin_amdgcn_wmma_f32_16x16x4_f32(false, a, false, bb,
                                                (short)0, acc, false, false);
  }

  const float* SQb = SQ + (size_t)b * NN;
  float* Db = D2 + (size_t)b * NN * NN;
  float lsum = 0.f;
#pragma unroll
  for (int r = 0; r < 8; ++r) {
    int gm = r0 + kh * 8 + r;   // C/D layout: lanes16-31 hold M = 8+r
    int gn = c0 + m;
    float d2  = SQb[gm] + SQb[gn] - 2.f * acc[r];
    float d2c = fmaxf(d2, 0.f);
    Db[(size_t)gm * NN + gn] = d2c;
    lsum += sqrtf(d2c);
  }
#pragma unroll
  for (int off = 16; off; off >>= 1) lsum += __shfl_xor(lsum, off, 32);
  if (lane == 0) atomicAdd(sig, lsum);
}

__global__ void k_sigmafin(float* __restrict__ sig) {
  float sigma = sig[0] / (float)((size_t)BB * NN * NN);
  sig[1] = 1.f / (2.f * sigma * sigma + 1e-30f);
}

// one wave per row: w = exp(-d2c*inv), iterative top-16 via wave argmax, scatter into A
__global__ void k_topk(const float* __restrict__ D2, const float* __restrict__ sig,
                       float* __restrict__ A) {
  int row = blockIdx.x;
  int b = row >> 10, n = row & 1023;
  int lane = threadIdx.x;
  float inv = sig[1];
  const float* Drow = D2 + ((size_t)b * NN + n) * NN;

  float v[32];
#pragma unroll
  for (int j = 0; j < 32; ++j) v[j] = __expf(-Drow[lane + 32 * j] * inv);

  float* Ab = A + (size_t)b * NN * NN;
  for (int r = 0; r < KNN; ++r) {
    float bv = -1.f;
    int bj = 0;
#pragma unroll
    for (int j = 0; j < 32; ++j) {
      if (v[j] > bv) { bv = v[j]; bj = j; }
    }
    int gc = bj * 32 + lane;            // == global column
#pragma unroll
    for (int off = 16; off; off >>= 1) {
      float ov = __shfl_xor(bv, off, 32);
      int   og = __shfl_xor(gc, off, 32);
      if (ov > bv || (ov == bv && og < gc)) { bv = ov; gc = og; }
    }
    if (lane == (gc & 31)) {            // winner lane retires its slot
      int wj = gc >> 5;
#pragma unroll
      for (int j = 0; j < 32; ++j)
        if (j == wj) v[j] = -1.f;
    }
    if (lane == 0) {
      Ab[(size_t)n * NN + gc] = bv;     // symmetric scatter (w is symmetric)
      Ab[(size_t)gc * NN + n] = bv;
    }
  }
}

__global__ void k_deg(const float* __restrict__ A, float* __restrict__ Dis) {
  int row = blockIdx.x;
  int b = row >> 10, n = row & 1023;
  int lane = threadIdx.x;
  const float* Ar = A + ((size_t)b * NN + n) * NN;
  float s = 0.f;
  for (int m = lane; m < NN; m += 32) s += Ar[m];
#pragma unroll
  for (int off = 16; off; off >>= 1) s += __shfl_xor(s, off, 32);
  if (lane == 0) Dis[(size_t)b * NN + n] = rsqrtf(fmaxf(s, 1e-12f));
}

// null eigenvector of L: v0 ∝ sqrt(deg) = 1/Dis, normalized per batch
__global__ void k_v0(const float* __restrict__ Dis, float* __restrict__ V0) {
  __shared__ float red[256];
  int b = blockIdx.x, tid = threadIdx.x;
  const float* db = Dis + (size_t)b * NN;
  float s = 0.f;
  for (int i = tid; i < NN; i += 256) { float d = db[i]; s += 1.f / (d * d); }
  red[tid] = s; __syncthreads();
  for (int off = 128; off; off >>= 1) {
    if (tid < off) red[tid] += red[tid + off];
    __syncthreads();
  }
  float inv = rsqrtf(red[0]);
  for (int i = tid; i < NN; i += 256) V0[(size_t)b * NN + i] = inv / db[i];
}

// L = I - Dis A Dis, written in place over A
__global__ void k_buildL(float* __restrict__ A, const float* __restrict__ Dis) {
  size_t idx = (size_t)blockIdx.x * 256 + threadIdx.x;
  size_t total = (size_t)BB * NN * NN;
  size_t stride = (size_t)gridDim.x * 256;
  for (; idx < total; idx += stride) {
    size_t r = idx % ((size_t)NN * NN);
    int b = (int)(idx / ((size_t)NN * NN));
    int n = (int)(r / NN), m = (int)(r % NN);
    float a = A[idx];
    A[idx] = ((n == m) ? 1.f : 0.f) - Dis[(size_t)b * NN + n] * a * Dis[(size_t)b * NN + m];
  }
}

__global__ void k_initV(float* __restrict__ U) {
  int idx = blockIdx.x * 256 + threadIdx.x;
  if (idx >= BB * NV * NN) return;
  uint32_t h = (uint32_t)idx * 2654435761u;
  h ^= h >> 16; h *= 2246822519u; h ^= h >> 13;
  U[idx] = (float)(h & 0xFFFFFFu) * (1.f / 16777216.f) - 0.5f;
}

// dst[j] = alpha*src[j] + beta*(L src[j]) ; one wave per (b,row), all NV columns at once
__global__ void k_matvec(const float* __restrict__ Lmat, const float* __restrict__ src,
                         float* __restrict__ dst, float alpha, float beta) {
  int gid  = blockIdx.x * 8 + (threadIdx.x >> 5);
  int lane = threadIdx.x & 31;
  int b = gid >> 10, n = gid & 1023;
  const float* Lrow = Lmat + ((size_t)b * NN + n) * NN;
  const float* S = src + (size_t)b * NV * NN;
  float a0 = 0.f, a1 = 0.f, a2 = 0.f, a3 = 0.f;
  for (int m = lane; m < NN; m += 32) {
    if (m + 256 < NN) __builtin_prefetch(Lrow + m + 256, 0, 0);
    float lv = Lrow[m];
    a0 += lv * S[0 * NN + m];
    a1 += lv * S[1 * NN + m];
    a2 += lv * S[2 * NN + m];
    a3 += lv * S[3 * NN + m];
  }
#pragma unroll
  for (int off = 16; off; off >>= 1) {
    a0 += __shfl_xor(a0, off, 32);
    a1 += __shfl_xor(a1, off, 32);
    a2 += __shfl_xor(a2, off, 32);
    a3 += __shfl_xor(a3, off, 32);
  }
  if (lane == 0) {
    float* Dp = dst + (size_t)b * NV * NN;
    Dp[0 * NN + n] = alpha * S[0 * NN + n] + beta * a0;
    Dp[1 * NN + n] = alpha * S[1 * NN + n] + beta * a1;
    Dp[2 * NN + n] = alpha * S[2 * NN + n] + beta * a2;
    Dp[3 * NN + n] = alpha * S[3 * NN + n] + beta * a3;
  }
}

// modified Gram-Schmidt: project out v0, orthonormalize NV columns (one block per batch)
__global__ void k_mgs(const float* __restrict__ srcAll, float* __restrict__ dstAll,
                      const float* __restrict__ V0All) {
  __shared__ float wbuf[NN];
  __shared__ float red[256];
  int b = blockIdx.x, tid = threadIdx.x;
  const float* src = srcAll + (size_t)b * NV * NN;
  float* dst = dstAll + (size_t)b * NV * NN;
  const float* v0 = V0All + (size_t)b * NN;
  for (int j = 0; j < NV; ++j) {
    for (int i = tid; i < NN; i += 256) wbuf[i] = src[(size_t)j * NN + i];
    __syncthreads();
    float d0 = blockDot256(wbuf, v0, red);
    for (int i = tid; i < NN; i += 256) wbuf[i] -= d0 * v0[i];
    __syncthreads();
    for (int jj = 0; jj < j; ++jj) {
      float dji = blockDot256(wbuf, dst + (size_t)jj * NN, red);
      for (int i = tid; i < NN; i += 256) wbuf[i] -= dji * dst[(size_t)jj * NN + i];
      __syncthreads();
    }
    float nn2 = blockDot256(wbuf, wbuf, red);
    float sc = (nn2 > 1e-30f) ? rsqrtf(nn2) : 0.f;
    for (int i = tid; i < NN; i += 256) dst[(size_t)j * NN + i] = wbuf[i] * sc;
    __syncthreads();
  }
}

// Rayleigh-Ritz: T = U^T (L U), 4x4 Jacobi, sort ascending, rotate, sign-fix
__global__ void k_ritz(const float* __restrict__ Uall, const float* __restrict__ LUall,
                       float* __restrict__ SM) {
  __shared__ float red[256];
  __shared__ float Tsh[16];
  __shared__ float Qs[16];
  int b = blockIdx.x, tid = threadIdx.x;
  const float* Ub  = Uall  + (size_t)b * NV * NN;
  const float* LUb = LUall + (size_t)b * NV * NN;

  for (int i = 0; i < NV; ++i)
    for (int j = 0; j < NV; ++j) {
      float d = blockDot256(Ub + (size_t)i * NN, LUb + (size_t)j * NN, red);
      if (tid == 0) Tsh[i * 4 + j] = d;
    }
  __syncthreads();

  if (tid == 0) {
    float Tm[4][4], Qm[4][4];
    for (int i = 0; i < 4; ++i)
      for (int j = 0; j < 4; ++j) {
        Tm[i][j] = 0.5f * (Tsh[i * 4 + j] + Tsh[j * 4 + i]);
        Qm[i][j] = (i == j) ? 1.f : 0.f;
      }
    for (int sweep = 0; sweep < 12; ++sweep)
      for (int p = 0; p < 3; ++p)
        for (int q = p + 1; q < 4; ++q) {
          float apq = Tm[p][q];
          if (fabsf(apq) < 1e-12f) continue;
          float theta = 0.5f * (Tm[q][q] - Tm[p][p]) / apq;
          float t = ((theta >= 0.f) ? 1.f : -1.f) /
                    (fabsf(theta) + sqrtf(theta * theta + 1.f));
          float c = rsqrtf(t * t + 1.f);
          float s = t * c;
          for (int k = 0; k < 4; ++k) {
            float akp = Tm[k][p], akq = Tm[k][q];
            Tm[k][p] = c * akp - s * akq;
            Tm[k][q] = s * akp + c * akq;
          }
          for (int k = 0; k < 4; ++k) {
            float apk = Tm[p][k], aqk = Tm[q][k];
            Tm[p][k] = c * apk - s * aqk;
            Tm[q][k] = s * apk + c * aqk;
          }
          for (int k = 0; k < 4; ++k) {
            float qkp = Qm[k][p], qkq = Qm[k][q];
            Qm[k][p] = c * qkp - s * qkq;
            Qm[k][q] = s * qkp + c * qkq;
          }
        }
    int ordv[4] = {0, 1, 2, 3};
    for (int i = 0; i < 3; ++i) {
      int mn = i;
      for (int j = i + 1; j < 4; ++j)
        if (Tm[ordv[j]][ordv[j]] < Tm[ordv[mn]][ordv[mn]]) mn = j;
      int tmpi = ordv[i]; ordv[i] = ordv[mn]; ordv[mn] = tmpi;
    }
    for (int k = 0; k < 4; ++k)
      for (int j = 0; j < 4; ++j) Qs[j * 4 + k] = Qm[j][ordv[k]];
  }
  __syncthreads();

  for (int n = tid; n < NN; n += 256) {
    float u0 = Ub[0 * NN + n], u1 = Ub[1 * NN + n];
    float u2 = Ub[2 * NN + n], u3 = Ub[3 * NN + n];
    float sm[4];
#pragma unroll
    for (int k = 0; k < 4; ++k)
      sm[k] = u0 * Qs[0 * 4 + k] + u1 * Qs[1 * 4 + k] + u2 * Qs[2 * 4 + k] + u3 * Qs[3 * 4 + k];
    float mean = 0.25f * (sm[0] + sm[1] + sm[2] + sm[3]);
    float sg = (mean > 0.f) ? 1.f : ((mean < 0.f) ? -1.f : 0.f);
#pragma unroll
    for (int k = 0; k < 4; ++k)
      SM[((size_t)b * NN + n) * TK + k] = sm[k] * sg;
  }
}

// per-(batch,column) bitonic argsort of 1024 values in LDS; write both outputs
__global__ void k_sort(const float* __restrict__ SM, float* __restrict__ out) {
  __shared__ float sval[NN];
  __shared__ int   sidx[NN];
  int b = blockIdx.x >> 2;
  int k = blockIdx.x & 3;
  int tid = threadIdx.x;
  for (int i = tid; i < NN; i += 512) {
    sval[i] = SM[((size_t)b * NN + i) * TK + k];
    sidx[i] = i;
  }
  __syncthreads();
  for (int ks = 2; ks <= NN; ks <<= 1)
    for (int jj = ks >> 1; jj > 0; jj >>= 1) {
      for (int i = tid; i < NN; i += 512) {
        int ixj = i ^ jj;
        if (ixj > i) {
          bool up = ((i & ks) == 0);
          float a = sval[i], c = sval[ixj];
          bool sw = up ? (a > c) : (a < c);
          if (sw) {
            sval[i] = c; sval[ixj] = a;
            int ti = sidx[i]; sidx[i] = sidx[ixj]; sidx[ixj] = ti;
          }
        }
      }
      __syncthreads();
    }
  for (int i = tid; i < NN; i += 512) {
    out[((size_t)b * NN + i) * TK + k] = sval[i];
    out[(size_t)BB * NN * TK + ((size_t)b * NN + i) * TK + k] = (float)sidx[i];
  }
}

// ---------------- host entry ----------------
extern "C" void kernel_launch(void* const* d_in, const int* in_sizes, int n_in,
                              void* d_out, int out_size, void* d_ws, size_t ws_size,
                              hipStream_t stream) {
  (void)in_sizes; (void)n_in; (void)out_size; (void)ws_size;
  const float* F = (const float*)d_in[0];
  float* out = (float*)d_out;
  float* ws  = (float*)d_ws;

  float* X   = ws + offX;
  float* SQ  = ws + offSQ;
  float* SIG = ws + offSig;
  float* D2  = ws + offD2;
  float* A   = ws + offA;     // adjacency, then Laplacian in place
  float* DIS = ws + offDis;
  float* V0  = ws + offV0;
  float* U   = ws + offU;
  float* Wv  = ws + offWv;
  float* SM  = ws + offSM;

  k_zero<<<2048, 256, 0, stream>>>(A, (size_t)BB * NN * NN, SIG);
  k_transpose<<<(BB * DD_ * NN + 255) / 256, 256, 0, stream>>>(F, X);
  k_rownorm<<<(BB * NN + 255) / 256, 256, 0, stream>>>(X, SQ);
  k_gram<<<dim3(NN / 16, NN / 16, BB), 32, 0, stream>>>(X, SQ, D2, SIG);
  k_sigmafin<<<1, 1, 0, stream>>>(SIG);
  k_topk<<<BB * NN, 32, 0, stream>>>(D2, SIG, A);
  k_deg<<<BB * NN, 32, 0, stream>>>(A, DIS);
  k_v0<<<BB, 256, 0, stream>>>(DIS, V0);
  k_buildL<<<4096, 256, 0, stream>>>(A, DIS);
  k_initV<<<(BB * NV * NN + 255) / 256, 256, 0, stream>>>(U);
  k_mgs<<<BB, 256, 0, stream>>>(U, U, V0);
  for (int it = 0; it < NITER; ++it) {
    k_matvec<<<BB * NN / 8, 256, 0, stream>>>(A, U, Wv, 2.f, -1.f);  // (2I-L)U
    k_mgs<<<BB, 256, 0, stream>>>(Wv, U, V0);
  }
  k_matvec<<<BB * NN / 8, 256, 0, stream>>>(A, U, Wv, 0.f, 1.f);     // LU for Ritz
  k_ritz<<<BB, 256, 0, stream>>>(U, Wv, SM);
  k_sort<<<BB * TK, 512, 0, stream>>>(SM, out);
}